// GRU_8211977470408
// MI455X (gfx1250) — compile-verified
//
#include <hip/hip_runtime.h>

// Bidirectional GRU (B=4096, T=512, H=16, IN=1) + MLP head, fully fused.
// One wave (32 lanes) owns a 16-batch tile. Recurrent GEMM
//   gh^T = W_hh(48x16) x h^T(16x16)  done with V_WMMA_F32_16X16X4_F32:
//   3 gate tiles x 4 K-chunks = 12 WMMAs/step. h relaid out D->B via a
//   wave-private LDS tile (same-wave DS ops are in-order; no barriers).

typedef __attribute__((ext_vector_type(2))) float v2f;
typedef __attribute__((ext_vector_type(8))) float v8f;

#define T_LEN 512

__device__ __forceinline__ float fast_sigmoid(float x) {
    return __builtin_amdgcn_rcpf(1.0f + __expf(-x));
}
__device__ __forceinline__ float fast_tanh(float x) {
    return 2.0f * __builtin_amdgcn_rcpf(1.0f + __expf(-2.0f * x)) - 1.0f;
}
__device__ __forceinline__ v8f wmma_f32(v2f a, v2f b, v8f c) {
    // (neg_a, A, neg_b, B, c_mod, C, reuse_a, reuse_b)
    return __builtin_amdgcn_wmma_f32_16x16x4_f32(false, a, false, b, (short)0, c,
                                                 false, false);
}

__global__ __launch_bounds__(32) void gru_bidir_head_kernel(
    const float* __restrict__ x,
    const float* __restrict__ Wih_f, const float* __restrict__ Whh_f,
    const float* __restrict__ bih_f, const float* __restrict__ bhh_f,
    const float* __restrict__ Wih_b, const float* __restrict__ Whh_b,
    const float* __restrict__ bih_b, const float* __restrict__ bhh_b,
    const float* __restrict__ W1, const float* __restrict__ b1,
    const float* __restrict__ W2, const float* __restrict__ b2,
    float* __restrict__ out)
{
    (void)Whh_b; // backward dir starts from h0=0 -> gh == b_hh_b, no matmul
    __shared__ float lds[32 * 16]; // rows 0..15: h^T ; rows 16..31: h_bwd^T

    const int lane = threadIdx.x;   // 32 lanes, always convergent
    const int col  = lane & 15;     // batch column within tile / A-matrix row M
    const int hi   = lane >> 4;     // lane half
    const int bcol = blockIdx.x * 16 + col;

    // --- W_hh_f preloaded as WMMA A-operand chunks: A[gate_tile][k_chunk]
    // A(16x4 f32) layout: lane m<16: V0=(m,k0), V1=(m,k0+1); lane m+16: k0+2,k0+3
    v2f A[3][4];
#pragma unroll
    for (int t = 0; t < 3; ++t) {
        const float* row = Whh_f + (t * 16 + col) * 16;
#pragma unroll
        for (int kk = 0; kk < 4; ++kk) {
            v2f a;
            a.x = row[4 * kk + 2 * hi];
            a.y = row[4 * kk + 1 + 2 * hi];
            A[t][kk] = a;
        }
    }

    // Per-lane gate constants for the 8 rows this lane owns (M = r + 8*hi)
    float cR[8], cZ[8], cN[8], wr[8], wz[8], wn[8], bn[8], hD[8];
#pragma unroll
    for (int r = 0; r < 8; ++r) {
        const int j = r + 8 * hi;
        cR[r] = bhh_f[j]      + bih_f[j];        // fold both biases into C
        cZ[r] = bhh_f[16 + j] + bih_f[16 + j];
        cN[r] = bhh_f[32 + j];                    // n-gate: b_hh only (scaled by r)
        bn[r] = bih_f[32 + j];
        wr[r] = Wih_f[j];
        wz[r] = Wih_f[16 + j];
        wn[r] = Wih_f[32 + j];
        hD[r] = 0.0f;
        lds[j * 16 + col] = 0.0f;                 // h0 = 0 in the LDS tile
    }

    const float* xcol = x + (size_t)bcol * T_LEN;
    float4 xv = *(const float4*)xcol;             // software-pipelined x fetch

    for (int tc = 0; tc < T_LEN; tc += 4) {
        float xsa[4] = {xv.x, xv.y, xv.z, xv.w};
        if (tc + 4 < T_LEN) xv = *(const float4*)(xcol + tc + 4);
#pragma unroll
        for (int i = 0; i < 4; ++i) {
            const float xt = xsa[i];
            // h^T as B-operand chunks (4x16 f32): V0=(k0,n)|(k0+2,n), V1=+1
            v2f Bk[4];
#pragma unroll
            for (int kk = 0; kk < 4; ++kk) {
                v2f b;
                b.x = lds[(4 * kk + 2 * hi) * 16 + col];
                b.y = lds[(4 * kk + 1 + 2 * hi) * 16 + col];
                Bk[kk] = b;
            }
            v8f dr, dz, dn;
#pragma unroll
            for (int r = 0; r < 8; ++r) { dr[r] = cR[r]; dz[r] = cZ[r]; dn[r] = cN[r]; }
#pragma unroll
            for (int kk = 0; kk < 4; ++kk) {      // K=16 via 4 chained K=4 WMMAs
                dr = wmma_f32(A[0][kk], Bk[kk], dr);
                dz = wmma_f32(A[1][kk], Bk[kk], dz);
                dn = wmma_f32(A[2][kk], Bk[kk], dn);
            }
#pragma unroll
            for (int r = 0; r < 8; ++r) {
                const float rr = fast_sigmoid(dr[r] + xt * wr[r]);
                const float zz = fast_sigmoid(dz[r] + xt * wz[r]);
                const float nn = fast_tanh(xt * wn[r] + bn[r] + rr * dn[r]);
                const float hn = (1.0f - zz) * nn + zz * hD[r];
                hD[r] = hn;
                lds[(r + 8 * hi) * 16 + col] = hn; // D-layout -> LDS for next step
            }
        }
    }

    // Backward direction: exactly one step from h0=0 on x[:, T-1]; gh = b_hh_b.
    const float xl = xcol[T_LEN - 1];
#pragma unroll
    for (int r = 0; r < 8; ++r) {
        const int j = r + 8 * hi;
        const float rr = fast_sigmoid(xl * Wih_b[j]      + bih_b[j]      + bhh_b[j]);
        const float zz = fast_sigmoid(xl * Wih_b[16 + j] + bih_b[16 + j] + bhh_b[16 + j]);
        const float nn = fast_tanh   (xl * Wih_b[32 + j] + bih_b[32 + j] + rr * bhh_b[32 + j]);
        lds[(16 + j) * 16 + col] = (1.0f - zz) * nn;   // h0=0 -> z*h term vanishes
    }

    // Head layer 1: hid^T = sigmoid( W1(16x32) x last^T(32x16) + b1 ), via WMMA
    v8f hv;
#pragma unroll
    for (int r = 0; r < 8; ++r) hv[r] = b1[r + 8 * hi];
    const float* w1row = W1 + col * 32;
#pragma unroll
    for (int kk = 0; kk < 8; ++kk) {
        v2f a, b;
        a.x = w1row[4 * kk + 2 * hi];
        a.y = w1row[4 * kk + 1 + 2 * hi];
        b.x = lds[(4 * kk + 2 * hi) * 16 + col];
        b.y = lds[(4 * kk + 1 + 2 * hi) * 16 + col];
        hv = wmma_f32(a, b, hv);
    }
#pragma unroll
    for (int r = 0; r < 8; ++r)
        lds[(r + 8 * hi) * 16 + col] = fast_sigmoid(hv[r]); // hid^T into rows 0..15

    // Head layer 2 (16 -> 2): lane half = output column
    float acc = b2[hi];
#pragma unroll
    for (int j = 0; j < 16; ++j)
        acc += W2[hi * 16 + j] * lds[j * 16 + col];
    out[(size_t)bcol * 2 + hi] = fast_sigmoid(acc);
}

extern "C" void kernel_launch(void* const* d_in, const int* in_sizes, int n_in,
                              void* d_out, int out_size, void* d_ws, size_t ws_size,
                              hipStream_t stream) {
    (void)n_in; (void)out_size; (void)d_ws; (void)ws_size;
    const float* x     = (const float*)d_in[0];
    const float* Wih_f = (const float*)d_in[1];
    const float* Whh_f = (const float*)d_in[2];
    const float* bih_f = (const float*)d_in[3];
    const float* bhh_f = (const float*)d_in[4];
    const float* Wih_b = (const float*)d_in[5];
    const float* Whh_b = (const float*)d_in[6];
    const float* bih_b = (const float*)d_in[7];
    const float* bhh_b = (const float*)d_in[8];
    const float* W1    = (const float*)d_in[9];
    const float* b1    = (const float*)d_in[10];
    const float* W2    = (const float*)d_in[11];
    const float* b2    = (const float*)d_in[12];
    float* out = (float*)d_out;

    const int B = in_sizes[0] / T_LEN;            // 4096
    gru_bidir_head_kernel<<<dim3(B / 16), dim3(32), 0, stream>>>(
        x, Wih_f, Whh_f, bih_f, bhh_f, Wih_b, Whh_b, bih_b, bhh_b,
        W1, b1, W2, b2, out);
}